// Expand_19670950216228
// MI455X (gfx1250) — compile-verified
//
#include <hip/hip_runtime.h>

#define B_    16
#define S_    1024
#define D_    512
#define TPAD_ 8192

typedef float v4f __attribute__((ext_vector_type(4)));
typedef int   v4i __attribute__((ext_vector_type(4)));

typedef __attribute__((address_space(1))) v4i* g4ptr_t;  // global int4*
typedef __attribute__((address_space(3))) v4i* l4ptr_t;  // LDS int4*

__device__ __forceinline__ g4ptr_t to_global4(const void* p) {
    // generic -> u64 -> addrspace(1): identity mapping for global pointers
    return (g4ptr_t)(unsigned long long)p;
}
__device__ __forceinline__ l4ptr_t to_lds4(const void* p) {
    // generic LDS addr low 32 bits == LDS offset (aperture mapping), valid AS(3) value
    return (l4ptr_t)(unsigned int)(unsigned long long)p;
}

__device__ __forceinline__ void wait_async0() {
#if __has_builtin(__builtin_amdgcn_s_wait_asynccnt)
    __builtin_amdgcn_s_wait_asynccnt(0);
#else
    asm volatile("s_wait_asynccnt 0" ::: "memory");
#endif
}

#if __has_builtin(__builtin_amdgcn_global_load_async_to_lds_b128) && \
    __has_builtin(__builtin_amdgcn_global_store_async_from_lds_b128)
#define USE_ASYNC_LDS 1
#else
#define USE_ASYNC_LDS 0
#endif

// ---------------- Kernel A: per-batch inclusive scan of repeat counts -------
// Wave32 shuffle scan (5 steps) + scan of 32 wave sums; only 2 barriers
// instead of 20 (latency matters: only B_=16 blocks exist for this stage).
__global__ __launch_bounds__(S_) void scan_kernel(const float* __restrict__ dims,
                                                  int* __restrict__ ends) {
    const int b    = blockIdx.x;
    const int tid  = threadIdx.x;          // 0..1023
    const int lane = tid & 31;
    const int wave = tid >> 5;             // 0..31

    int val = (int)(dims[b * S_ + tid] + 0.5f);   // exact: integer-valued floats

    // Intra-wave inclusive scan
    #pragma unroll
    for (int off = 1; off < 32; off <<= 1) {
        int n = __shfl_up(val, off, 32);
        if (lane >= off) val += n;
    }

    __shared__ int wsum[32];
    if (lane == 31) wsum[wave] = val;
    __syncthreads();

    if (wave == 0) {
        int s = wsum[lane];
        #pragma unroll
        for (int off = 1; off < 32; off <<= 1) {
            int n = __shfl_up(s, off, 32);
            if (lane >= off) s += n;
        }
        wsum[lane] = s;                    // exclusive-of-next-wave prefix (inclusive here)
    }
    __syncthreads();

    const int offset = (wave == 0) ? 0 : wsum[wave - 1];
    ends[b * S_ + tid] = val + offset;     // inclusive cumsum == reference `ends`
}

// ---------------- Kernel B: scatter each source row to its d output rows ----
// Block = one (b,s) row. 128 threads x 16B = 2KB row. Each wave32 owns a
// disjoint 512B slice of the LDS buffer -> only wave-local asynccnt waits.
__global__ __launch_bounds__(128) void expand_kernel(const float* __restrict__ x,
                                                     const int* __restrict__ ends,
                                                     float* __restrict__ out) {
    const int s   = blockIdx.x;
    const int b   = blockIdx.y;
    const int tid = threadIdx.x;

    const int e  = ends[b * S_ + s];
    const int st = (s == 0) ? 0 : ends[b * S_ + s - 1];
    const int d  = e - st;                       // 1..8

    const float* src = x + ((size_t)(b * S_ + s)) * D_ + (size_t)tid * 4;
    float* dst0      = out + ((size_t)b * TPAD_ + (size_t)st) * D_ + (size_t)tid * 4;

#if USE_ASYNC_LDS
    __shared__ __align__(16) char buf[128 * 16];
    // Stage row into LDS via the async data mover (x read from HBM exactly once).
    __builtin_amdgcn_global_load_async_to_lds_b128(to_global4(src),
                                                   to_lds4(buf + tid * 16),
                                                   0, /*cpol*/0);
    wait_async0();   // async LDS writes may complete out of order vs async reads
    // Stream the row out d times straight from LDS, nontemporal (write-once output).
    for (int r = 0; r < d; ++r) {
        __builtin_amdgcn_global_store_async_from_lds_b128(to_global4(dst0 + (size_t)r * D_),
                                                          to_lds4(buf + tid * 16),
                                                          0, /*cpol: TH=NT*/1);
    }
    wait_async0();
#else
    v4f v = *(const v4f*)src;
    for (int r = 0; r < d; ++r) {
        __builtin_nontemporal_store(v, (v4f*)(dst0 + (size_t)r * D_));
    }
#endif
}

// ---------------- Kernel C: zero-fill padded tail rows ----------------------
__global__ __launch_bounds__(256) void zero_tail(const int* __restrict__ ends,
                                                 float* __restrict__ out) {
    const int b     = blockIdx.y;
    const int total = ends[b * S_ + (S_ - 1)];
    const int rowbase = blockIdx.x * 16;
    if (rowbase + 16 <= total) return;           // fully inside valid region
    const int tid = threadIdx.x;
    v4f z = {0.0f, 0.0f, 0.0f, 0.0f};
    // 16 rows * 128 float4 per row, strided over 256 threads
    for (int i = tid; i < 16 * 128; i += 256) {
        const int row = rowbase + (i >> 7);
        if (row >= total) {
            __builtin_nontemporal_store(
                z, (v4f*)(out + ((size_t)b * TPAD_ + (size_t)row) * D_ + (size_t)(i & 127) * 4));
        }
    }
}

extern "C" void kernel_launch(void* const* d_in, const int* in_sizes, int n_in,
                              void* d_out, int out_size, void* d_ws, size_t ws_size,
                              hipStream_t stream) {
    const float* x    = (const float*)d_in[0];   // (B,S,D) f32
    const float* dims = (const float*)d_in[1];   // (B,S)  f32
    float* out = (float*)d_out;                  // (B,T_PAD,D) f32
    int* ends  = (int*)d_ws;                     // B*S ints = 64KB scratch

    scan_kernel<<<dim3(B_), dim3(S_), 0, stream>>>(dims, ends);
    expand_kernel<<<dim3(S_, B_), dim3(128), 0, stream>>>(x, ends, out);
    zero_tail<<<dim3(TPAD_ / 16, B_), dim3(256), 0, stream>>>(ends, out);
}